// NBodyGNN_40931038331300
// MI455X (gfx1250) — compile-verified
//
#include <hip/hip_runtime.h>

typedef __attribute__((ext_vector_type(2))) float v2f;
typedef __attribute__((ext_vector_type(8))) float v8f;

#define FH   128   // hidden width (GEMM output width for all 3 layers)
#define SWST (FH + 8)  // sW LDS row stride: 136 -> conflict-free B-fragment reads
#define TPB  5     // row-tiles processed per GEMM block (W staged once per block)

// ---------------------------------------------------------------- utilities
__global__ void gcn_zero(float* __restrict__ p, int n) {
    int i = blockIdx.x * blockDim.x + threadIdx.x;
    if (i < n) p[i] = 0.0f;
}

// degree: one thread per (edge or self-loop), unit weight
__global__ void gcn_deg(const int* __restrict__ dst, float* __restrict__ deg,
                        int E, int M /* = E+N */) {
    int e = blockIdx.x * blockDim.x + threadIdx.x;
    if (e >= M) return;
    int d = (e < E) ? dst[e] : (e - E);
    atomicAdd(&deg[d], 1.0f);
}

__global__ void gcn_dinv(const float* __restrict__ deg, float* __restrict__ dinv, int N) {
    int i = blockIdx.x * blockDim.x + threadIdx.x;
    if (i >= N) return;
    float dg = deg[i];
    dinv[i] = (dg > 0.0f) ? __builtin_amdgcn_rsqf(dg) : 0.0f;
}

// ---------------------------------------------------------------- WMMA GEMM
// out[N x 128] = A[N x K] @ W[K x 128] using V_WMMA_F32_16X16X4_F32.
// Block = 256 threads = 8 waves; each wave owns a 16-col slab.
// Block is persistent over TPB consecutive 16-row tiles; W staged in LDS once.
__global__ void __launch_bounds__(256)
gcn_gemm_wmma(const float* __restrict__ A, const float* __restrict__ W,
              float* __restrict__ out, int K, int Kpad, int N) {
    extern __shared__ float lds[];
    float* sW = lds;                        // Kpad * SWST
    float* sA = lds + (size_t)Kpad * SWST;  // 16 * (Kpad+4)
    const int sAstride = Kpad + 4;

    const int tid  = threadIdx.x;
    const int wave = tid >> 5;              // 0..7 -> 16-col slab
    const int lane = tid & 31;              // wave32
    const int half = lane >> 4;             // 0/1
    const int m    = lane & 15;
    const int colBase = wave * 16;

    // stage W once (pad rows >= K with zero); row stride SWST in LDS
    for (int idx = tid; idx < Kpad * FH; idx += 256) {
        int r = idx >> 7, c = idx & (FH - 1);
        sW[r * SWST + c] = (r < K) ? W[r * FH + c] : 0.0f;
    }

    for (int t = 0; t < TPB; ++t) {
        const int rowBase = (blockIdx.x * TPB + t) * 16;
        if (rowBase >= N) break;

        __syncthreads();   // sA safe to overwrite (also orders first sW stage)

        // stage 16 x K tile of A
        if (K == FH) {
            // contiguous 16x128 block: vector b128 loads
            const float4* src4 = reinterpret_cast<const float4*>(A + (size_t)rowBase * FH);
            for (int idx = tid; idx < 16 * FH / 4; idx += 256) {
                int r = (idx * 4) >> 7, c = (idx * 4) & (FH - 1);
                *reinterpret_cast<float4*>(&sA[r * sAstride + c]) = src4[idx];
            }
        } else {
            for (int idx = tid; idx < 16 * Kpad; idx += 256) {
                int r = idx / Kpad, c = idx - r * Kpad;
                int row = rowBase + r;
                sA[r * sAstride + c] = (c < K && row < N) ? A[(size_t)row * K + c] : 0.0f;
            }
        }

        // prefetch next row-tile of A (global_prefetch_b8; fire-and-forget)
        const int nextRow = rowBase + 16;
        if (t + 1 < TPB && nextRow < N && tid < 32) {
            __builtin_prefetch(A + ((size_t)nextRow + (tid & 15)) * K + (tid >> 4) * 64, 0, 0);
        }

        __syncthreads();

        v8f acc = {};
        for (int k = 0; k < Kpad; k += 4) {
            const int ka = k + 2 * half;    // lanes 0-15: K=k,k+1 ; lanes 16-31: K=k+2,k+3
            v2f a, b;
            a.x = sA[m * sAstride + ka];
            a.y = sA[m * sAstride + ka + 1];
            b.x = sW[ka * SWST + colBase + m];
            b.y = sW[(ka + 1) * SWST + colBase + m];
            acc = __builtin_amdgcn_wmma_f32_16x16x4_f32(false, a, false, b,
                                                        (short)0, acc, false, false);
        }

        // D layout: acc[r] = D[r + 8*half][colBase + m]
        #pragma unroll
        for (int r = 0; r < 8; ++r) {
            int row = rowBase + r + 8 * half;
            if (row < N) out[(size_t)row * FH + colBase + m] = acc[r];
        }
    }
}

// ---------------------------------------------------------------- scatter-add
// One wave per (edge | self-loop): msg = T[src]*norm, atomically added to G[dst].
// Each lane handles 4 consecutive features (32 lanes * 4 = 128).
__global__ void __launch_bounds__(256)
gcn_scatter(const float* __restrict__ T, const float* __restrict__ dinv,
            const int* __restrict__ src, const int* __restrict__ dst,
            float* __restrict__ G, int E, int M) {
    int e = blockIdx.x * 8 + (threadIdx.x >> 5);
    if (e >= M) return;
    int lane = threadIdx.x & 31;
    int s, d;
    if (e < E) { s = src[e]; d = dst[e]; }
    else       { s = e - E;  d = s;      }
    float nrm = dinv[s] * dinv[d];
    const float4 v = *reinterpret_cast<const float4*>(T + (size_t)s * FH + lane * 4);
    float* gp = G + (size_t)d * FH + lane * 4;
    atomicAdd(gp + 0, v.x * nrm);
    atomicAdd(gp + 1, v.y * nrm);
    atomicAdd(gp + 2, v.z * nrm);
    atomicAdd(gp + 3, v.w * nrm);
}

__global__ void gcn_bias_relu(float* __restrict__ H, const float* __restrict__ b, int total) {
    int i = blockIdx.x * blockDim.x + threadIdx.x;
    if (i >= total) return;
    H[i] = fmaxf(H[i] + b[i & (FH - 1)], 0.0f);
}

// out[N x 3] = H[N x 128] @ Wout[128 x 3] + bout
__global__ void gcn_out(const float* __restrict__ H, const float* __restrict__ Wout,
                        const float* __restrict__ bout, float* __restrict__ out, int N) {
    int idx = blockIdx.x * blockDim.x + threadIdx.x;
    int node = idx / 3, j = idx - node * 3;
    if (node >= N) return;
    float acc = bout[j];
    const float* h = H + (size_t)node * FH;
    #pragma unroll 4
    for (int k = 0; k < FH; ++k) acc = fmaf(h[k], Wout[k * 3 + j], acc);
    out[(size_t)node * 3 + j] = acc;
}

// ---------------------------------------------------------------- launcher
extern "C" void kernel_launch(void* const* d_in, const int* in_sizes, int n_in,
                              void* d_out, int out_size, void* d_ws, size_t ws_size,
                              hipStream_t stream) {
    const float* x     = (const float*)d_in[0];   // N x 7
    const int*   ei    = (const int*)  d_in[1];   // 2 x E (row-major: src then dst)
    const float* W1    = (const float*)d_in[2];
    const float* b1    = (const float*)d_in[3];
    const float* W2    = (const float*)d_in[4];
    const float* b2    = (const float*)d_in[5];
    const float* W3    = (const float*)d_in[6];
    const float* b3    = (const float*)d_in[7];
    const float* Wout  = (const float*)d_in[8];
    const float* bout  = (const float*)d_in[9];
    float* out = (float*)d_out;

    const int FIN = 7;
    const int N = in_sizes[0] / FIN;
    const int E = in_sizes[1] / 2;
    const int M = E + N;                   // edges + self loops
    const int* srcI = ei;
    const int* dstI = ei + E;

    float* ws   = (float*)d_ws;
    float* deg  = ws;                      // N
    float* dinv = ws + N;                  // N
    float* H    = ws + 2 * (size_t)N;      // N x 128
    float* T    = H  + (size_t)N * FH;     // N x 128

    const int BT = 256;
    dim3 blk(BT);
    const int nTiles = (N + 15) / 16;
    const int gN    = (N + BT - 1) / BT;
    const int gM    = (M + BT - 1) / BT;
    const int gNF   = (N * FH + BT - 1) / BT;
    const int gScat = (M + 7) / 8;
    const int gGemm = (nTiles + TPB - 1) / TPB;
    const int gOut  = (N * 3 + BT - 1) / BT;

    // normalization
    gcn_zero<<<gN, blk, 0, stream>>>(deg, N);
    gcn_deg <<<gM, blk, 0, stream>>>(dstI, deg, E, M);
    gcn_dinv<<<gN, blk, 0, stream>>>(deg, dinv, N);

    // layer 1: K = 7 (padded to 8)
    {
        int K = FIN, Kpad = 8;
        size_t ldsB = ((size_t)Kpad * SWST + 16 * (Kpad + 4)) * sizeof(float);
        gcn_gemm_wmma<<<gGemm, blk, ldsB, stream>>>(x, W1, T, K, Kpad, N);
        gcn_zero   <<<gNF, blk, 0, stream>>>(H, N * FH);
        gcn_scatter<<<gScat, blk, 0, stream>>>(T, dinv, srcI, dstI, H, E, M);
        gcn_bias_relu<<<gNF, blk, 0, stream>>>(H, b1, N * FH);
    }
    // layers 2,3: K = 128
    const float* Ws[2] = {W2, W3};
    const float* bs[2] = {b2, b3};
    for (int l = 0; l < 2; ++l) {
        int K = FH, Kpad = FH;
        size_t ldsB = ((size_t)Kpad * SWST + 16 * (Kpad + 4)) * sizeof(float);
        gcn_gemm_wmma<<<gGemm, blk, ldsB, stream>>>(H, Ws[l], T, K, Kpad, N);
        gcn_zero   <<<gNF, blk, 0, stream>>>(H, N * FH);
        gcn_scatter<<<gScat, blk, 0, stream>>>(T, dinv, srcI, dstI, H, E, M);
        gcn_bias_relu<<<gNF, blk, 0, stream>>>(H, bs[l], N * FH);
    }

    // output projection
    gcn_out<<<gOut, blk, 0, stream>>>(H, Wout, bout, out, N);
}